// GNN_57372173140422
// MI455X (gfx1250) — compile-verified
//
#include <hip/hip_runtime.h>
#include <hip/hip_bf16.h>

typedef __attribute__((ext_vector_type(2))) float v2f;
typedef __attribute__((ext_vector_type(8))) float v8f;

// ---------------------------------------------------------------------------
// CSR construction: histogram, dinv, scan, placement
// ---------------------------------------------------------------------------
__global__ void k_zero_i32(int* __restrict__ p, int n) {
    int i = blockIdx.x * blockDim.x + threadIdx.x;
    if (i < n) p[i] = 0;
}

__global__ void k_hist(const int* __restrict__ ei, int* __restrict__ cnt, int nE) {
    int e = blockIdx.x * blockDim.x + threadIdx.x;
    if (e < nE) atomicAdd(&cnt[ei[nE + e]], 1);  // dst row
}

__global__ void k_dinv(const int* __restrict__ cnt, float* __restrict__ dinv, int n) {
    int i = blockIdx.x * blockDim.x + threadIdx.x;
    if (i < n) dinv[i] = rsqrtf((float)cnt[i] + 1.0f);  // +1 self-loop; always > 0
}

// per-block exclusive scan (256 wide, Hillis-Steele in LDS)
__global__ __launch_bounds__(256) void k_scan_block(const int* __restrict__ cnt,
                                                    int* __restrict__ off,
                                                    int* __restrict__ bsum, int n) {
    __shared__ int s[256];
    const int tid = threadIdx.x;
    const int i = blockIdx.x * 256 + tid;
    int v = (i < n) ? cnt[i] : 0;
    s[tid] = v;
    __syncthreads();
    for (int d = 1; d < 256; d <<= 1) {
        int t = (tid >= d) ? s[tid - d] : 0;
        __syncthreads();
        s[tid] += t;
        __syncthreads();
    }
    if (i < n) off[i] = s[tid] - v;  // exclusive within block
    if (tid == 255) bsum[blockIdx.x] = s[255];
}

// single-block exclusive scan of per-block sums (nb <= 256)
__global__ __launch_bounds__(256) void k_scan_sums(const int* __restrict__ bsum,
                                                   int* __restrict__ boff, int nb) {
    __shared__ int s[256];
    const int tid = threadIdx.x;
    int v = (tid < nb) ? bsum[tid] : 0;
    s[tid] = v;
    __syncthreads();
    for (int d = 1; d < 256; d <<= 1) {
        int t = (tid >= d) ? s[tid - d] : 0;
        __syncthreads();
        s[tid] += t;
        __syncthreads();
    }
    if (tid < nb) boff[tid] = s[tid] - v;
}

__global__ void k_scan_add(int* __restrict__ off, int* __restrict__ cur,
                           const int* __restrict__ boff, int n) {
    int i = blockIdx.x * blockDim.x + threadIdx.x;
    if (i < n) {
        int o = off[i] + boff[i >> 8];
        off[i] = o;
        cur[i] = o;  // running cursor for placement; ends at off+cnt (= row end)
    }
}

__global__ void k_place(const int* __restrict__ ei, int* __restrict__ cur,
                        int* __restrict__ csr, int nE) {
    int e = blockIdx.x * blockDim.x + threadIdx.x;
    if (e < nE) {
        int d = ei[nE + e];
        int p = atomicAdd(&cur[d], 1);
        csr[p] = ei[e];  // src
    }
}

// ---------------------------------------------------------------------------
// WMMA GEMM: D[M x NCOLS] = A[M x 128] @ W[128 x NCOLS]  (fp32, full precision)
// One wave computes a 16 x NCOLS strip. W staged in LDS in fragment order so
// each B fragment is one 8B-aligned ds_load_b64 into an adjacent VGPR pair.
// Fragment layouts per CDNA5 ISA 7.12.2 (wave32):
//   A 16x4 : lane m=lane%16, half=lane/16 -> (A[m][k+2h], A[m][k+2h+1])
//   B 4x16 : col n=lane%16               -> (W[k+2h][n],  W[k+2h+1][n])
//   D 16x16: vgpr r -> row r+8h, col n=lane%16
// ---------------------------------------------------------------------------
template <int NCOLS>
__global__ __launch_bounds__(256) void k_gemm_wmma(const float* __restrict__ A,
                                                   const float* __restrict__ W,
                                                   float* __restrict__ D,
                                                   int mtiles) {
    constexpr int K  = 128;
    constexpr int NT = NCOLS / 16;
    __shared__ float sB[K * NCOLS];  // [(k4*NT + t)*32 + lane] * 2 floats

    for (int slot = threadIdx.x; slot < 32 * NT * 32; slot += 256) {
        int k4  = slot / (NT * 32);
        int rem = slot - k4 * (NT * 32);
        int t   = rem >> 5;
        int ln  = rem & 31;
        int r0  = k4 * 4 + 2 * (ln >> 4);
        int n   = t * 16 + (ln & 15);
        sB[slot * 2 + 0] = W[r0 * NCOLS + n];
        sB[slot * 2 + 1] = W[(r0 + 1) * NCOLS + n];
    }
    __syncthreads();

    const int wave = threadIdx.x >> 5;
    const int tile = blockIdx.x * 8 + wave;
    if (tile >= mtiles) return;  // wave-uniform; EXEC stays all-ones below

    const int lane = threadIdx.x & 31;
    const int half = lane >> 4;
    const int mr   = lane & 15;
    const int m0   = tile * 16;

    v8f acc[NT] = {};
    const float* arow = A + (size_t)(m0 + mr) * K;

#pragma unroll
    for (int k4 = 0; k4 < 32; ++k4) {
        v2f a = *(const v2f*)(arow + k4 * 4 + 2 * half);
#pragma unroll
        for (int t = 0; t < NT; ++t) {
            v2f b = *(const v2f*)&sB[((k4 * NT + t) * 32 + lane) * 2];
            acc[t] = __builtin_amdgcn_wmma_f32_16x16x4_f32(
                false, a, false, b, (short)0, acc[t], false, false);
        }
    }

#pragma unroll
    for (int t = 0; t < NT; ++t) {
#pragma unroll
        for (int r = 0; r < 8; ++r) {
            D[(size_t)(m0 + r + 8 * half) * NCOLS + t * 16 + mr] = acc[t][r];
        }
    }
}

// ---------------------------------------------------------------------------
// Gather aggregation, one wave per node (C = 32*VEC channels, VEC/lane):
//   out[i] = [relu]( bias + dinv[i]^2 * h[i] + sum_e dinv[i]*dinv[src]*h[src] )
// All gathers hit L2 (h fits easily in 192 MB); single coalesced store.
// ---------------------------------------------------------------------------
template <int VEC, bool RELU>
__global__ __launch_bounds__(256) void k_aggregate(const float* __restrict__ h,
                                                   const int* __restrict__ csr,
                                                   const int* __restrict__ off,
                                                   const int* __restrict__ rowend,
                                                   const float* __restrict__ dinv,
                                                   const float* __restrict__ bias,
                                                   float* __restrict__ out, int n) {
    const int node = (blockIdx.x * 256 + threadIdx.x) >> 5;
    const int lane = threadIdx.x & 31;
    if (node >= n) return;  // wave-uniform
    constexpr int C = 32 * VEC;

    const float di = dinv[node];
    const float self = di * di;
    float acc[VEC];

    {   // bias + self-loop
        const float* hp = h + (size_t)node * C + lane * VEC;
        if constexpr (VEC == 4) {
            float4 v = *(const float4*)hp;
            float4 bv = *(const float4*)&bias[lane * 4];
            acc[0] = bv.x + self * v.x; acc[1] = bv.y + self * v.y;
            acc[2] = bv.z + self * v.z; acc[3] = bv.w + self * v.w;
        } else {
            float2 v = *(const float2*)hp;
            float2 bv = *(const float2*)&bias[lane * 2];
            acc[0] = bv.x + self * v.x; acc[1] = bv.y + self * v.y;
        }
    }

    const int e1 = rowend[node];
    for (int e = off[node]; e < e1; ++e) {
        int s = csr[e];
        float nrm = di * dinv[s];
        const float* sp = h + (size_t)s * C + lane * VEC;
        if constexpr (VEC == 4) {
            float4 v = *(const float4*)sp;
            acc[0] += nrm * v.x; acc[1] += nrm * v.y;
            acc[2] += nrm * v.z; acc[3] += nrm * v.w;
        } else {
            float2 v = *(const float2*)sp;
            acc[0] += nrm * v.x; acc[1] += nrm * v.y;
        }
    }

#pragma unroll
    for (int j = 0; j < VEC; ++j)
        if constexpr (RELU) acc[j] = fmaxf(acc[j], 0.0f);

    float* op = out + (size_t)node * C + lane * VEC;
    if constexpr (VEC == 4) {
        float4 o; o.x = acc[0]; o.y = acc[1]; o.z = acc[2]; o.w = acc[3];
        *(float4*)op = o;
    } else {
        float2 o; o.x = acc[0]; o.y = acc[1];
        *(float2*)op = o;
    }
}

// ---------------------------------------------------------------------------
extern "C" void kernel_launch(void* const* d_in, const int* in_sizes, int n_in,
                              void* d_out, int out_size, void* d_ws, size_t ws_size,
                              hipStream_t stream) {
    const float* x  = (const float*)d_in[0];
    const int*   ei = (const int*)d_in[1];   // [2, E] flat: src row then dst row
    const float* W1 = (const float*)d_in[2];
    const float* b1 = (const float*)d_in[3];
    const float* W2 = (const float*)d_in[4];
    const float* b2 = (const float*)d_in[5];
    float* out = (float*)d_out;

    const int IN_C = 128, HID_C = 128, OUT_C = 64;
    const int N  = in_sizes[0] / IN_C;  // 50000
    const int nE = in_sizes[1] / 2;     // 800000

    // workspace layout (all pieces 16B-multiple sized -> float4 alignment holds)
    float* dinv = (float*)d_ws;                  // [N]
    int*   cnt  = (int*)(dinv + N);              // [N]
    int*   off  = cnt + N;                       // [N]
    int*   cur  = off + N;                       // [N] cursor; ends as row-end
    int*   bsum = cur + N;                       // [256]
    int*   boff = bsum + 256;                    // [256]
    int*   csr  = boff + 256;                    // [nE] src ids grouped by dst
    float* h    = (float*)(csr + nE);            // [N,128]; reused as g [N,64]
    float* agg  = h + (size_t)N * HID_C;         // [N,128] relu(agg1+b1)

    const int T = 256;
    const int nb = (N + 255) / 256;  // 196 <= 256

    // ---- CSR by dst + normalization ----
    k_zero_i32<<<(N + T - 1) / T, T, 0, stream>>>(cnt, N);
    k_hist<<<(nE + T - 1) / T, T, 0, stream>>>(ei, cnt, nE);
    k_dinv<<<(N + T - 1) / T, T, 0, stream>>>(cnt, dinv, N);
    k_scan_block<<<nb, T, 0, stream>>>(cnt, off, bsum, N);
    k_scan_sums<<<1, T, 0, stream>>>(bsum, boff, nb);
    k_scan_add<<<(N + T - 1) / T, T, 0, stream>>>(off, cur, boff, N);
    k_place<<<(nE + T - 1) / T, T, 0, stream>>>(ei, cur, csr, nE);

    const int mtiles  = N / 16;          // 3125
    const int gblocks = (mtiles + 7) / 8;
    const int ablocks = ((N * 32) + T - 1) / T;  // one wave per node

    // ---- layer 1: h = x@W1 ; agg = relu(bias + self + gathered sum) ----
    k_gemm_wmma<128><<<gblocks, T, 0, stream>>>(x, W1, h, mtiles);
    k_aggregate<4, true><<<ablocks, T, 0, stream>>>(h, csr, off, cur, dinv, b1, agg, N);

    // ---- layer 2: g = agg@W2 ; out = bias + self + gathered sum ----
    k_gemm_wmma<64><<<gblocks, T, 0, stream>>>(agg, W2, h /* g */, mtiles);
    k_aggregate<2, false><<<ablocks, T, 0, stream>>>(h, csr, off, cur, dinv, b2, out, N);
}